// RotationNetwork_81484119540140
// MI455X (gfx1250) — compile-verified
//
#include <hip/hip_runtime.h>

// ---------------------------------------------------------------------------
// RotationNetwork: v = silu(x) @ W^T + b  ->  skew(A) -> expm(A)
//   B=8192, N=1024, D=64, K=2016
// Strategy: full-f32 WMMA (V_WMMA_F32_16X16X4_F32) everywhere.
//   Kernel 1: tiled GEMM with silu fused on the A loads (fast rcp sigmoid).
//   Kernel 2: per-batch expm via scaling-and-squaring + Taylor (Horner),
//             all 64x64x64 matmuls in LDS via WMMA tiles.
// ---------------------------------------------------------------------------

typedef float v2f __attribute__((ext_vector_type(2)));
typedef float v8f __attribute__((ext_vector_type(8)));

#define BATCH 8192
#define NFEAT 1024
#define DDIM  64
#define KOUT  2016   // D*(D-1)/2
#define MEXP  12     // Taylor degree
#define SQR   6      // squarings (||A||2 ~ 12 -> ||X|| ~ 0.19)

__device__ __forceinline__ v8f wmma4(v2f a, v2f b, v8f c) {
  // D = A(16x4,f32) * B(4x16,f32) + C(16x16,f32)
  return __builtin_amdgcn_wmma_f32_16x16x4_f32(
      /*neg_a=*/false, a, /*neg_b=*/false, b,
      /*c_mod=*/(short)0, c, /*reuse_a=*/false, /*reuse_b=*/false);
}

// silu via v_exp + v_rcp (1-ulp HW reciprocal) -- avoids the ~10-op IEEE
// division expansion (v_div_scale/v_div_fmas/v_div_fixup) in the inner loop.
__device__ __forceinline__ float silu_f(float z) {
  float e = __expf(-z);
  return z * __builtin_amdgcn_rcpf(1.0f + e);
}

// ---------------------------------------------------------------------------
// Kernel 1: v[8192, 2016] = silu(x[8192,1024]) @ W[2016,1024]^T + b[2016]
// Grid: (126, 64) blocks of 256 threads (8 waves). Wave w in block (nx, my)
// computes output tile (mt = my*8 + w, nt = nx). All 8 waves share the same
// W tile -> L0/L2 reuse; x is L2-resident (32MB << 192MB L2).
// ---------------------------------------------------------------------------
__global__ __launch_bounds__(256) void gemm_silu_wmma(
    const float* __restrict__ x, const float* __restrict__ W,
    const float* __restrict__ bias, float* __restrict__ v)
{
  const int lane = threadIdx.x & 31;
  const int wave = threadIdx.x >> 5;
  const int nt   = blockIdx.x;              // 0..125
  const int mt   = blockIdx.y * 8 + wave;   // 0..511
  const int hi   = lane >> 4;               // 0: lanes 0-15, 1: lanes 16-31
  const int l16  = lane & 15;
  const int klo  = hi * 2;                  // K sub-offset per A/B layout

  const int mrow = mt * 16 + l16;           // A-fragment row (M)
  const int ncol = nt * 16 + l16;           // B-fragment col (N)

  const float* __restrict__ xrow = x + (size_t)mrow * NFEAT + klo;
  const float* __restrict__ wrow = W + (size_t)ncol * NFEAT + klo; // W^T col = W row

  v8f acc = {};
#pragma unroll 8
  for (int kk = 0; kk < NFEAT; kk += 4) {
    // A frag: {silu(x[m][kk+klo]), silu(x[m][kk+klo+1])}
    float2 xa = *(const float2*)(xrow + kk);
    // B frag: {W^T[kk+klo][n], W^T[kk+klo+1][n]} = contiguous in W's row
    float2 wb = *(const float2*)(wrow + kk);
    v2f a; a.x = silu_f(xa.x); a.y = silu_f(xa.y);
    v2f b; b.x = wb.x;         b.y = wb.y;
    acc = wmma4(a, b, acc);
  }

  const float bb = bias[ncol];
#pragma unroll
  for (int r = 0; r < 8; ++r) {
    // C/D layout: VGPR r holds M = r (lanes 0-15) / M = r+8 (lanes 16-31)
    const int row = mt * 16 + r + hi * 8;
    v[(size_t)row * KOUT + ncol] = acc[r] + bb;
  }
}

// ---------------------------------------------------------------------------
// 64x64x64 matmul in LDS: Cd = A @ B * invk + addI * I
// 16 output tiles (4x4 of 16x16), 8 waves -> 2 tiles per wave, 16 k-steps.
// ---------------------------------------------------------------------------
__device__ __forceinline__ void mm64_wmma(const float* __restrict__ A,
                                          const float* __restrict__ B,
                                          float* __restrict__ Cd,
                                          int lane, int wave,
                                          float invk, float addI)
{
  const int hi  = lane >> 4;
  const int l16 = lane & 15;
  const int klo = hi * 2;

  for (int t = wave; t < 16; t += 8) {
    const int mtile = (t >> 2) * 16;
    const int ntile = (t & 3) * 16;
    const float* __restrict__ arow = A + (mtile + l16) * DDIM + klo;
    const float* __restrict__ bcol = B + klo * DDIM + (ntile + l16);
    v8f acc = {};
#pragma unroll
    for (int kk = 0; kk < DDIM; kk += 4) {
      v2f a;  a.x = arow[kk];         a.y = arow[kk + 1];
      v2f bf; bf.x = bcol[kk * DDIM]; bf.y = bcol[kk * DDIM + DDIM];
      acc = wmma4(a, bf, acc);
    }
#pragma unroll
    for (int r = 0; r < 8; ++r) {
      const int row = mtile + r + hi * 8;
      const int col = ntile + l16;
      Cd[row * DDIM + col] = acc[r] * invk + ((row == col) ? addI : 0.0f);
    }
  }
}

// ---------------------------------------------------------------------------
// Kernel 2: per-batch expm of skew-symmetric A built from v[b, :].
// X = A * 2^-SQR; T = sum_{j<=MEXP} X^j/j! via Horner; then square SQR times.
// One block (256 threads / 8 waves) per batch element. LDS: 3 x 16KB buffers.
// ---------------------------------------------------------------------------
__global__ __launch_bounds__(256) void expm_wmma(
    const float* __restrict__ v, float* __restrict__ out)
{
  __shared__ float Xs[DDIM * DDIM];
  __shared__ float Ta[DDIM * DDIM];
  __shared__ float Tb[DDIM * DDIM];

  const int b    = blockIdx.x;
  const int tid  = threadIdx.x;
  const int lane = tid & 31;
  const int wave = tid >> 5;
  const float scale = 1.0f / (float)(1 << SQR);

  // Compile-time 1/k table: keeps IEEE division out of the Horner loop.
  const float invk_tab[MEXP] = {
    0.0f, 1.0f, 1.0f/2.0f, 1.0f/3.0f, 1.0f/4.0f, 1.0f/5.0f,
    1.0f/6.0f, 1.0f/7.0f, 1.0f/8.0f, 1.0f/9.0f, 1.0f/10.0f, 1.0f/11.0f
  };

  const float* __restrict__ vb = v + (size_t)b * KOUT;

  // Build X (scaled skew matrix) and T = I + X/MEXP (first Horner step folded in).
  for (int p = tid; p < DDIM * DDIM; p += 256) {
    const int i = p >> 6, j = p & 63;
    float val = 0.0f;
    if (i < j) {
      const int idx = i * 63 - (i * (i - 1)) / 2 + (j - i - 1);
      val = vb[idx] * scale;
    } else if (i > j) {
      const int idx = j * 63 - (j * (j - 1)) / 2 + (i - j - 1);
      val = -vb[idx] * scale;
    }
    Xs[p] = val;
    Ta[p] = val * (1.0f / (float)MEXP) + ((i == j) ? 1.0f : 0.0f);
  }
  __syncthreads();

  float* cur = Ta;
  float* oth = Tb;

  // Horner: T <- I + (X @ T) / k, k = MEXP-1 .. 1
  for (int k = MEXP - 1; k >= 1; --k) {
    mm64_wmma(Xs, cur, oth, lane, wave, invk_tab[k], 1.0f);
    __syncthreads();
    float* tmp = cur; cur = oth; oth = tmp;
  }

  // Squaring: T <- T @ T, SQR times
  for (int s = 0; s < SQR; ++s) {
    mm64_wmma(cur, cur, oth, lane, wave, 1.0f, 0.0f);
    __syncthreads();
    float* tmp = cur; cur = oth; oth = tmp;
  }

  float* __restrict__ ob = out + (size_t)b * (DDIM * DDIM);
  for (int p = tid; p < DDIM * DDIM; p += 256) ob[p] = cur[p];
}

// ---------------------------------------------------------------------------
extern "C" void kernel_launch(void* const* d_in, const int* in_sizes, int n_in,
                              void* d_out, int out_size, void* d_ws, size_t ws_size,
                              hipStream_t stream) {
  const float* x    = (const float*)d_in[0];  // (8192, 1024) f32
  const float* W    = (const float*)d_in[1];  // (2016, 1024) f32
  const float* bias = (const float*)d_in[2];  // (2016,) f32
  float* out = (float*)d_out;                 // (8192, 64, 64) f32
  float* v   = (float*)d_ws;                  // (8192, 2016) f32 scratch (~63MB)

  dim3 g1(KOUT / 16, BATCH / (16 * 8));       // (126, 64)
  gemm_silu_wmma<<<g1, 256, 0, stream>>>(x, W, bias, v);
  expm_wmma<<<BATCH, 256, 0, stream>>>(v, out);
}